// Conv2d_new_9199819948497
// MI455X (gfx1250) — compile-verified
//
#include <hip/hip_runtime.h>

// Conv2d 3x3 s1 p1, NCHW fp32, B=16 Cin=16 Cout=16 H=W=512.
// Implicit GEMM on CDNA5 WMMA f32 16x16x4, K = Cin*9 = 144 in 36 steps.
//
// K-order trick: features permuted as (g = kh*3+kw major, cin minor) with
// cin assigned per WMMA slot as cin = 4j + 2*half + v. Combined with
// cell-major LDS layout xs[(r*66+c)*18 + cin] and swizzled weights
// wl[m*148 + g*16 + cin], every fragment address is base + compile-time
// immediate -> one ds_load_b64 (pair-merged to ds_load_2addr_b64) per
// fragment, no per-step VALU.
//
// 8-row blocks: halo reload factor 10/8 = 1.25x -> ~0.61 GB total HBM
// traffic (~26us roofline at 23.3 TB/s).

#define HH   512
#define WW   512
#define CIN  16
#define COUT 16

#define XP     18    // padded cin pitch per (row,col) cell (conflict-free, even)
#define HALO_C 66    // 64 output cols + 2 halo
#define HALO_R 10    // 8 output rows + 2 halo
#define WPITCH 148   // padded per-cout weight pitch (conflict-free, even)

typedef __attribute__((ext_vector_type(2))) float v2f;
typedef __attribute__((ext_vector_type(8))) float v8f;

// Workgroup: 256 threads = 8 wave32. Covers batch b, 8 output rows, 64 cols,
// all 16 couts. Wave w owns output row ro = w: 4 tiles of 16 px.
__global__ __launch_bounds__(256) void conv3x3_wmma_f32(
    const float* __restrict__ x, const float* __restrict__ wgt,
    const float* __restrict__ bias, float* __restrict__ out)
{
  __shared__ float xs[HALO_R * HALO_C * XP]; // [r][c][cin], pitch 18 (47,520 B)
  __shared__ float wl[COUT * WPITCH];        // [cout][g*16+cin]      ( 9,472 B)

  const int tid   = threadIdx.x;
  const int bid   = blockIdx.x;
  const int strip = bid & 7;          // 8 strips of 64 cols
  const int hblk  = (bid >> 3) & 63;  // 64 blocks of 8 rows
  const int b     = bid >> 9;         // batch
  const int c0    = strip * 64;
  const int h0    = hblk * 8;

  // ---- weights -> LDS: read OIHW coalesced, write swizzled [m][g*16+cin] ----
  for (int i = tid; i < COUT * 144; i += 256) {
    const int m   = i / 144;
    const int rem = i - m * 144;
    const int cin = rem / 9;
    const int g   = rem - cin * 9;     // kh*3 + kw
    wl[m * WPITCH + g * 16 + cin] = wgt[i];
  }

  // ---- input halo -> LDS: coalesced global rows, cell-major LDS writes ----
  const float* xb = x + (size_t)b * CIN * HH * WW;
  for (int i = tid; i < CIN * HALO_R * HALO_C; i += 256) {
    const int c   = i % HALO_C;
    const int t   = i / HALO_C;
    const int r   = t % HALO_R;
    const int cin = t / HALO_R;
    const int gh  = h0 - 1 + r;
    const int gc  = c0 - 1 + c;
    float v = 0.0f;
    if ((unsigned)gh < (unsigned)HH && (unsigned)gc < (unsigned)WW)
      v = xb[(size_t)(cin * HH + gh) * WW + gc];
    xs[(r * HALO_C + c) * XP + cin] = v;
  }
  __syncthreads();

  const int lane = tid & 31;
  const int wave = tid >> 5;
  const int half = lane >> 4;      // lanes 16-31 hold K+2 / M+8 per ISA layout
  const int col  = lane & 15;      // pixel within tile / cout row M
  const int ro   = wave;           // output row within block (0..7)

  // Loop-invariant per-lane bases (element indices); all per-step offsets are
  // compile-time immediates after full unroll.
  const int Db = (ro * HALO_C + col) * XP + 2 * half;
  const int Dw = col * WPITCH + 2 * half;

  v8f acc[4] = {v8f{}, v8f{}, v8f{}, v8f{}};

#pragma unroll
  for (int g = 0; g < 9; ++g) {
    const int kh = g / 3;
    const int kw = g - kh * 3;
    const int bcell = (kh * HALO_C + kw) * XP;   // compile-time constant
#pragma unroll
    for (int j = 0; j < 4; ++j) {
      // A: W[m][cin = 4j+2h+v, g]  -> contiguous pair, one ds_load_b64
      v2f a = *(const v2f*)&wl[Dw + g * 16 + 4 * j];
#pragma unroll
      for (int t = 0; t < 4; ++t) {
        // B: X[r=ro+kh][c = t*16+col+kw][cin = 4j+2h+v] -> contiguous pair
        v2f bf = *(const v2f*)&xs[Db + bcell + t * 16 * XP + 4 * j];
        acc[t] = __builtin_amdgcn_wmma_f32_16x16x4_f32(
            false, a, false, bf, (short)0, acc[t], false, false);
      }
    }
  }

  // ---- store: C/D layout -> cout = j + 8*half, pixel = col ----
  const int gh = h0 + ro;
#pragma unroll
  for (int t = 0; t < 4; ++t) {
    const int gc = c0 + t * 16 + col;
#pragma unroll
    for (int j = 0; j < 8; ++j) {
      const int cout = j + half * 8;
      out[((size_t)(b * COUT + cout) * HH + gh) * WW + gc] = acc[t][j] + bias[cout];
    }
  }
}

extern "C" void kernel_launch(void* const* d_in, const int* in_sizes, int n_in,
                              void* d_out, int out_size, void* d_ws, size_t ws_size,
                              hipStream_t stream) {
  (void)in_sizes; (void)n_in; (void)d_ws; (void)ws_size; (void)out_size;
  const float* x    = (const float*)d_in[0];
  const float* wgt  = (const float*)d_in[1];
  const float* bias = (const float*)d_in[2];
  float* out        = (float*)d_out;

  // grid = batch(16) * row-blocks(64) * col-strips(8) = 8192 workgroups
  dim3 grid(16 * (HH / 8) * (WW / 64));
  conv3x3_wmma_f32<<<grid, 256, 0, stream>>>(x, wgt, bias, out);
}